// GNNMultiMotion_76768245449449
// MI455X (gfx1250) — compile-verified
//
#include <hip/hip_runtime.h>
#include <hip/hip_bf16.h>

#define NN 4096
#define XDIM 8
#define HDIM 128
#define NHEADS 4
#define QKV (HDIM * NHEADS)      // 512
#define E_TOT 135168             // 131072 + 4096
#define DEC_HID 256
#define DEC_OUT 372              // 6*30*2 + 2*6
#define MASK_WORDS (NN * (NN / 32))  // 524288
#define NEG_INF (-__builtin_inff())

typedef __attribute__((ext_vector_type(2))) float v2f;
typedef __attribute__((ext_vector_type(8))) float v8f;

// D(16x16 f32) = A(16x4 f32) * B(4x16 f32) + C
__device__ __forceinline__ v8f wmma4(v2f a, v2f b, v8f c) {
    return __builtin_amdgcn_wmma_f32_16x16x4_f32(
        false, a, false, b, (short)0, c, false, false);
}

__device__ __forceinline__ v2f ldpair(const float* p) {
    v2f r; r.x = p[0]; r.y = p[1]; return r;
}

// ---------------- node encoder: Hn = relu(relu(X@Wx1+b1)@Wx2+b2) -------------
__global__ void encoder_kernel(const float* __restrict__ X,
                               const float* __restrict__ Wx1, const float* __restrict__ bx1,
                               const float* __restrict__ Wx2, const float* __restrict__ bx2,
                               float* __restrict__ Hn) {
    int n = blockIdx.x;
    int j = threadIdx.x;           // 0..127
    __shared__ float xr[XDIM];
    __shared__ float h1[HDIM];
    if (j < XDIM) xr[j] = X[n * XDIM + j];
    __syncthreads();
    float acc = bx1[j];
#pragma unroll
    for (int i = 0; i < XDIM; ++i) acc += xr[i] * Wx1[i * HDIM + j];
    h1[j] = fmaxf(acc, 0.f);
    __syncthreads();
    float acc2 = bx2[j];
#pragma unroll 16
    for (int i = 0; i < HDIM; ++i) acc2 += h1[i] * Wx2[i * HDIM + j];
    Hn[n * HDIM + j] = fmaxf(acc2, 0.f);
}

// ---------------- adjacency bitmask ------------------------------------------
__global__ void mask_clear_kernel(unsigned int* __restrict__ mask) {
    int i = blockIdx.x * blockDim.x + threadIdx.x;
    if (i < MASK_WORDS) mask[i] = 0u;
}

__global__ void mask_scatter_kernel(const int* __restrict__ EI,
                                    unsigned int* __restrict__ mask) {
    int e = blockIdx.x * blockDim.x + threadIdx.x;
    if (e >= E_TOT) return;
    int src = EI[e];             // query row n
    int dst = EI[E_TOT + e];     // key col m
    atomicOr(&mask[src * (NN / 32) + (dst >> 5)], 1u << (dst & 31));
}

// ---------------- weight transpose: Bt[n*K + k] = B[k*N + n] -----------------
__global__ void transpose_kernel(const float* __restrict__ B, float* __restrict__ Bt,
                                 int K, int N) {
    int i = blockIdx.x * blockDim.x + threadIdx.x;
    if (i >= K * N) return;
    int k = i / N, n = i - k * N;
    Bt[(size_t)n * K + k] = B[i];
}

// ---------------- fp32 WMMA GEMM, 16x64 strip per wave, sw-pipelined ---------
// C = act(A[M,K] @ B[K,NOUT] + bias + resid), B given transposed (Bt[NOUT,K]).
// STORE_T: write C^T at C[(size_t)col*M + row] instead. K, NOUT compile-time.
template <int K, int NOUT, bool STORE_T>
__global__ void __launch_bounds__(256, 1)
gemm_wmma_kernel(const float* __restrict__ A,
                 const float* __restrict__ Bt,
                 const float* __restrict__ bias,
                 const float* __restrict__ resid,
                 float* __restrict__ C,
                 int M, int do_relu) {
    int wave = threadIdx.x >> 5;
    int lane = threadIdx.x & 31;
    int strip = blockIdx.x * (blockDim.x >> 5) + wave;
    const int stripsN = NOUT >> 6;       // 64-wide strips
    int tm = strip / stripsN;
    int tn = strip % stripsN;
    if (tm * 16 >= M) return;

    int half = lane >> 4;
    int l15  = lane & 15;
    int kOff = half * 2;

    const float* arow = A + (size_t)(tm * 16 + l15) * K + kOff;
    const float* bt0  = Bt + (size_t)(tn * 64 + l15) * K + kOff;          // s=0
    const float* bt1  = bt0 + (size_t)16 * K;
    const float* bt2  = bt0 + (size_t)32 * K;
    const float* bt3  = bt0 + (size_t)48 * K;

    v8f c0 = {}, c1 = {}, c2 = {}, c3 = {};

    // software pipeline: loads for step k+4 issued before wmmas of step k
    v2f a  = ldpair(arow);
    v2f b0 = ldpair(bt0), b1 = ldpair(bt1), b2 = ldpair(bt2), b3 = ldpair(bt3);
#pragma unroll 8
    for (int k0 = 0; k0 < K; k0 += 4) {
        v2f an = a, b0n = b0, b1n = b1, b2n = b2, b3n = b3;
        const int k1 = k0 + 4;
        if (k1 < K) {                     // folds at compile time per iteration
            an  = ldpair(arow + k1);
            b0n = ldpair(bt0 + k1);
            b1n = ldpair(bt1 + k1);
            b2n = ldpair(bt2 + k1);
            b3n = ldpair(bt3 + k1);
        }
        c0 = wmma4(a, b0, c0);
        c1 = wmma4(a, b1, c1);
        c2 = wmma4(a, b2, c2);
        c3 = wmma4(a, b3, c3);
        a = an; b0 = b0n; b1 = b1n; b2 = b2n; b3 = b3n;
    }

    v8f cs[4] = {c0, c1, c2, c3};
#pragma unroll
    for (int s = 0; s < 4; ++s) {
        int col = tn * 64 + s * 16 + l15;
#pragma unroll
        for (int v = 0; v < 8; ++v) {
            int row = tm * 16 + v + 8 * half;
            float val = cs[s][v];
            if (bias)  val += bias[col];
            if (resid) val += resid[(size_t)row * NOUT + col];
            if (do_relu) val = fmaxf(val, 0.f);
            if (STORE_T) C[(size_t)col * M + row] = val;
            else         C[(size_t)row * NOUT + col] = val;
        }
    }
}

// ---------------- masked multi-head attention (flash style) ------------------
// One wave handles (head h, 16-query-row tile tm); loops all 256 key tiles.
// V is provided transposed: vt[(h*128+d)][m], m contiguous -> b64 fragment loads.
// launch_bounds(128,1): register-fat kernel; forbid spilling of qa/outc.
__global__ void __launch_bounds__(128, 1)
gat_attention_kernel(const float* __restrict__ q,
                     const float* __restrict__ kmat,
                     const float* __restrict__ vt,
                     const unsigned int* __restrict__ maskbits,
                     float* __restrict__ attn) {
    __shared__ float smem[4 * 256];                 // 4 waves * 16x16 alpha tile
    int wave = threadIdx.x >> 5;
    int lane = threadIdx.x & 31;
    int wg = blockIdx.x * 4 + wave;                 // 0..1023
    int h  = wg >> 8;                               // head 0..3
    int tm = wg & 255;                              // query tile 0..255
    int half = lane >> 4;
    int l15  = lane & 15;
    int kOff = half * 2;
    int wbase = wave * 256;
    const float SCALE = 0.08838834764831845f;       // 1/sqrt(128)

    // Q fragments (pre-scaled), row = tm*16 + l15, 32 K-steps of 4
    v2f qa[32];
    {
        const float* qrow = q + (size_t)(tm * 16 + l15) * QKV + h * HDIM + kOff;
#pragma unroll
        for (int kk = 0; kk < 32; ++kk) {
            qa[kk].x = qrow[kk * 4] * SCALE;
            qa[kk].y = qrow[kk * 4 + 1] * SCALE;
        }
    }

    v8f outc[8];
#pragma unroll
    for (int dt = 0; dt < 8; ++dt) outc[dt] = (v8f){};
    float rowMax[8], rowSum[8];
#pragma unroll
    for (int v = 0; v < 8; ++v) { rowMax[v] = NEG_INF; rowSum[v] = 0.f; }

    for (int tn = 0; tn < 256; ++tn) {
        // prefetch next key tile row (lowers to global_prefetch_b8)
        if (tn + 1 < 256) {
            __builtin_prefetch(kmat + (size_t)((tn + 1) * 16 + l15) * QKV + h * HDIM, 0, 1);
            __builtin_prefetch(vt + (size_t)(h * HDIM + l15 * 8) * NN + (tn + 1) * 16, 0, 1);
        }

        // ---- scores tile: S = Q * K^T (C-layout), 2-deep load pipeline ----
        v8f c = {};
        const float* krow = kmat + (size_t)(tn * 16 + l15) * QKV + h * HDIM + kOff;
        v2f kb0 = ldpair(krow);
        v2f kb1 = ldpair(krow + 4);
#pragma unroll
        for (int kk = 0; kk < 32; ++kk) {
            v2f kn = kb1;
            if (kk + 2 < 32) kn = ldpair(krow + (kk + 2) * 4);
            c = wmma4(qa[kk], kb0, c);
            kb0 = kb1; kb1 = kn;
        }

        // ---- mask + online softmax (row = v + 8*half across 16 lanes) ----
        // alpha written straight to LDS, accumulator rescaled in-loop:
        // keeps live ranges short (no p[]/scl[] arrays -> no spills).
        int colG = tn * 16 + l15;
#pragma unroll
        for (int v = 0; v < 8; ++v) {
            int rowG = tm * 16 + v + 8 * half;
            unsigned int w = maskbits[rowG * (NN / 32) + (colG >> 5)];
            float s = ((w >> (colG & 31)) & 1u) ? c[v] : NEG_INF;
            float t = s;
#pragma unroll
            for (int off = 1; off < 16; off <<= 1)
                t = fmaxf(t, __shfl_xor(t, off, 32));
            float mNew = fmaxf(rowMax[v], t);
            float sc, pv;
            if (mNew == NEG_INF) { sc = 1.f; pv = 0.f; }
            else { sc = __expf(rowMax[v] - mNew); pv = __expf(s - mNew); }
            float ps = pv;
#pragma unroll
            for (int off = 1; off < 16; off <<= 1)
                ps += __shfl_xor(ps, off, 32);
            rowSum[v] = rowSum[v] * sc + ps;
            rowMax[v] = mNew;
            smem[wbase + (v + 8 * half) * 16 + l15] = pv;   // alpha -> LDS
#pragma unroll
            for (int dt = 0; dt < 8; ++dt) outc[dt][v] *= sc;
        }

        // ---- O += alpha * V  (V^T: m contiguous), 1-deep load pipeline ----
        const float* vbase = vt + (size_t)(h * HDIM + l15) * NN + tn * 16 + kOff;
        // fragment address for (ks, dt): vbase + dt*16*NN + ks*4
        v2f vb = ldpair(vbase);             // idx 0: ks=0, dt=0
        v2f aAl = ldpair(&smem[wbase + l15 * 16 + kOff]);   // ks=0
#pragma unroll
        for (int idx = 0; idx < 32; ++idx) {
            int ks = idx >> 3;
            int dt = idx & 7;
            v2f vbn = vb;
            if (idx + 1 < 32) {
                int ks2 = (idx + 1) >> 3;
                int dt2 = (idx + 1) & 7;
                vbn = ldpair(vbase + (size_t)dt2 * 16 * NN + ks2 * 4);
            }
            outc[dt] = wmma4(aAl, vb, outc[dt]);
            if (dt == 7 && ks < 3)          // next ks: new alpha A-fragment
                aAl = ldpair(&smem[wbase + l15 * 16 + (ks + 1) * 4 + kOff]);
            vb = vbn;
        }
    }

    // ---- normalize and store [n, h*128 + d] ----
#pragma unroll
    for (int dt = 0; dt < 8; ++dt)
#pragma unroll
        for (int v = 0; v < 8; ++v) {
            int rowG = tm * 16 + v + 8 * half;
            int col  = dt * 16 + l15;
            float denom = rowSum[v];
            float val = (denom > 0.f) ? outc[dt][v] / denom : 0.f;
            attn[(size_t)rowG * QKV + h * HDIM + col] = val;
        }
}

// ---------------- decoder (target row only) ----------------------------------
__global__ void decoder_kernel(const float* __restrict__ Hn,
                               const int* __restrict__ tgt,
                               const float* __restrict__ Wd1, const float* __restrict__ bd1,
                               const float* __restrict__ Wd2, const float* __restrict__ bd2,
                               float* __restrict__ out) {
    __shared__ float ht[HDIM];
    __shared__ float h1[DEC_HID];
    int t = threadIdx.x;           // 0..255
    int target = tgt[0];
    if (t < HDIM) ht[t] = Hn[(size_t)target * HDIM + t];
    __syncthreads();
    float acc = bd1[t];
#pragma unroll 16
    for (int i = 0; i < HDIM; ++i) acc += ht[i] * Wd1[i * DEC_HID + t];
    h1[t] = fmaxf(acc, 0.f);
    __syncthreads();
    for (int j = t; j < DEC_OUT; j += DEC_HID) {
        float a = bd2[j];
        for (int i = 0; i < DEC_HID; ++i) a += h1[i] * Wd2[i * DEC_OUT + j];
        // raw layout: [mu(360) | logits(6) | log_var(6)]
        // output tuple order: mu, log_var, logits
        int o;
        if (j < 360)       o = j;        // mu
        else if (j < 366)  o = j + 6;    // logits -> after log_var
        else               o = j - 6;    // log_var -> right after mu
        out[o] = a;
    }
}

// -----------------------------------------------------------------------------
extern "C" void kernel_launch(void* const* d_in, const int* in_sizes, int n_in,
                              void* d_out, int out_size, void* d_ws, size_t ws_size,
                              hipStream_t stream) {
    (void)in_sizes; (void)n_in; (void)out_size; (void)ws_size;
    const float* X   = (const float*)d_in[0];
    const int*   EI  = (const int*)d_in[2];
    const int*   tgt = (const int*)d_in[3];
    const float* Wx1 = (const float*)d_in[4];
    const float* bx1 = (const float*)d_in[5];
    const float* Wx2 = (const float*)d_in[6];
    const float* bx2 = (const float*)d_in[7];
    // d_in[8..11] = edge encoder (dead code in reference) — skipped
    const float* Wq0 = (const float*)d_in[12];
    const float* Wk0 = (const float*)d_in[13];
    const float* Wv0 = (const float*)d_in[14];
    const float* Wp0 = (const float*)d_in[15];
    const float* bp0 = (const float*)d_in[16];
    const float* Wq1 = (const float*)d_in[17];
    const float* Wk1 = (const float*)d_in[18];
    const float* Wv1 = (const float*)d_in[19];
    const float* Wp1 = (const float*)d_in[20];
    const float* bp1 = (const float*)d_in[21];
    const float* Wd1 = (const float*)d_in[22];
    const float* bd1 = (const float*)d_in[23];
    const float* Wd2 = (const float*)d_in[24];
    const float* bd2 = (const float*)d_in[25];

    // workspace layout
    float* HnA = (float*)d_ws;                       // 4096*128
    float* HnB = HnA + (size_t)NN * HDIM;            // 4096*128
    unsigned int* mask = (unsigned int*)(HnB + (size_t)NN * HDIM);   // 524288 words
    float* q    = (float*)(mask + MASK_WORDS);       // 4096*512
    float* k    = q + (size_t)NN * QKV;              // 4096*512
    float* vt   = k + (size_t)NN * QKV;              // 512*4096 (V transposed)
    float* attn = vt + (size_t)NN * QKV;             // 4096*512
    float* wt   = attn + (size_t)NN * QKV;           // 512*128 weight transpose buf

    const int WELEMS = HDIM * QKV;                   // 65536 (all gemm weights)
    const int TP_BLOCKS = (WELEMS + 255) / 256;

    // encoder + mask
    encoder_kernel<<<NN, HDIM, 0, stream>>>(X, Wx1, bx1, Wx2, bx2, HnA);
    mask_clear_kernel<<<(MASK_WORDS + 255) / 256, 256, 0, stream>>>(mask);
    mask_scatter_kernel<<<(E_TOT + 255) / 256, 256, 0, stream>>>(EI, mask);

    // strips: qkv gemm (M=4096,K=128,N=512): 256*8 strips / 8 waves = 256 blocks
    //         proj gemm (M=4096,K=512,N=128): 256*2 strips / 8 waves = 64 blocks
    const int QKV_BLOCKS  = (NN / 16) * (QKV / 64) / 8;
    const int PROJ_BLOCKS = (NN / 16) * (HDIM / 64) / 8;

    // ---- GAT layer 0 ----
    transpose_kernel<<<TP_BLOCKS, 256, 0, stream>>>(Wq0, wt, HDIM, QKV);
    gemm_wmma_kernel<HDIM, QKV, false><<<QKV_BLOCKS, 256, 0, stream>>>(HnA, wt, nullptr, nullptr, q, NN, 0);
    transpose_kernel<<<TP_BLOCKS, 256, 0, stream>>>(Wk0, wt, HDIM, QKV);
    gemm_wmma_kernel<HDIM, QKV, false><<<QKV_BLOCKS, 256, 0, stream>>>(HnA, wt, nullptr, nullptr, k, NN, 0);
    transpose_kernel<<<TP_BLOCKS, 256, 0, stream>>>(Wv0, wt, HDIM, QKV);
    gemm_wmma_kernel<HDIM, QKV, true><<<QKV_BLOCKS, 256, 0, stream>>>(HnA, wt, nullptr, nullptr, vt, NN, 0);
    gat_attention_kernel<<<256, 128, 0, stream>>>(q, k, vt, mask, attn);
    transpose_kernel<<<TP_BLOCKS, 256, 0, stream>>>(Wp0, wt, QKV, HDIM);
    gemm_wmma_kernel<QKV, HDIM, false><<<PROJ_BLOCKS, 256, 0, stream>>>(attn, wt, bp0, HnA, HnB, NN, 1);

    // ---- GAT layer 1 ----
    transpose_kernel<<<TP_BLOCKS, 256, 0, stream>>>(Wq1, wt, HDIM, QKV);
    gemm_wmma_kernel<HDIM, QKV, false><<<QKV_BLOCKS, 256, 0, stream>>>(HnB, wt, nullptr, nullptr, q, NN, 0);
    transpose_kernel<<<TP_BLOCKS, 256, 0, stream>>>(Wk1, wt, HDIM, QKV);
    gemm_wmma_kernel<HDIM, QKV, false><<<QKV_BLOCKS, 256, 0, stream>>>(HnB, wt, nullptr, nullptr, k, NN, 0);
    transpose_kernel<<<TP_BLOCKS, 256, 0, stream>>>(Wv1, wt, HDIM, QKV);
    gemm_wmma_kernel<HDIM, QKV, true><<<QKV_BLOCKS, 256, 0, stream>>>(HnB, wt, nullptr, nullptr, vt, NN, 0);
    gat_attention_kernel<<<256, 128, 0, stream>>>(q, k, vt, mask, attn);
    transpose_kernel<<<TP_BLOCKS, 256, 0, stream>>>(Wp1, wt, QKV, HDIM);
    gemm_wmma_kernel<QKV, HDIM, false><<<PROJ_BLOCKS, 256, 0, stream>>>(attn, wt, bp1, HnB, HnA, NN, 1);

    // ---- decoder ----
    decoder_kernel<<<1, DEC_HID, 0, stream>>>(HnA, tgt, Wd1, bd1, Wd2, bd2, (float*)d_out);
}